// SelfAttentionWithRelativePosition_37726992728847
// MI455X (gfx1250) — compile-verified
//
#include <hip/hip_runtime.h>

// ---------------------------------------------------------------------------
// SelfAttentionWithRelativePosition for MI455X (gfx1250, wave32, WMMA).
// B=1, L=384, DIM=256, HEADS=8, DIM_HEAD=32, POS_DIM=32.
// Algebraic restructuring: k/v = base(x_j) + pos_enc(i,j) @ W*_pos, so the
// O(L^2 * 288 * 256) reference GEMMs collapse to O(L * 256 * 256) base GEMMs
// plus per-pair 32-dim dot products. f32 WMMA (16x16x4) keeps full precision.
// ---------------------------------------------------------------------------

typedef __attribute__((ext_vector_type(2))) float v2f;
typedef __attribute__((ext_vector_type(8))) float v8f;

#define L_   384
#define DIM_ 256
#define H_   8
#define DH_  32
#define PI_F 3.14159265358979323846f

__device__ __forceinline__ v8f wmma4(v2f a, v2f b, v8f c) {
  // (neg_a, A, neg_b, B, c_mod, C, reuse_a, reuse_b)
  return __builtin_amdgcn_wmma_f32_16x16x4_f32(false, a, false, b, (short)0, c,
                                               false, false);
}

// ---------------------------------------------------------------------------
// Kernel 1: Qs = scale * x @ Wq ; Kb = x @ Wk[:256] ; Vb = x @ Wv[:256]
// grid = (48, 3), block = 256 (8 waves, 1 wave -> one 16x16 output tile)
// ---------------------------------------------------------------------------
__global__ __launch_bounds__(256) void qkv_gemm(
    const float* __restrict__ x, const float* __restrict__ Wq,
    const float* __restrict__ Wk, const float* __restrict__ Wv,
    float* __restrict__ Qs, float* __restrict__ Kb, float* __restrict__ Vb) {
  const int which = blockIdx.y;
  const float* W = (which == 0) ? Wq : (which == 1) ? Wk : Wv;
  float* Out = (which == 0) ? Qs : (which == 1) ? Kb : Vb;
  const float mul = (which == 0) ? 0.17677669529663687f : 1.0f;  // 32^-0.5

  const int wave = threadIdx.x >> 5, lane = threadIdx.x & 31;
  const int t = blockIdx.x * 8 + wave;   // 0..383 tiles (24 x 16)
  const int tm = t >> 4, tn = t & 15;
  const int r16 = lane & 15;
  const int kh = (lane >> 4) << 1;       // 0 or 2
  const int arow = tm * 16 + r16;
  const int bcol = tn * 16 + r16;

  v8f acc = {};
  for (int k0 = 0; k0 < DIM_; k0 += 4) {
    v2f a = *(const v2f*)(x + arow * DIM_ + k0 + kh);
    v2f b;
    b.x = W[(k0 + kh) * DIM_ + bcol];
    b.y = W[(k0 + kh + 1) * DIM_ + bcol];
    acc = wmma4(a, b, acc);
  }
  const int rbase = (lane >> 4) * 8;
#pragma unroll
  for (int r = 0; r < 8; ++r)
    Out[(tm * 16 + rbase + r) * DIM_ + tn * 16 + r16] = acc[r] * mul;
}

// ---------------------------------------------------------------------------
// Kernel 2: Acoef[h,i,p] = sum_d Qs[i, h*32+d] * Wk[256+p, h*32+d]
// per head: [384,32] @ [32,32].  grid = 48, block = 256.
// ---------------------------------------------------------------------------
__global__ __launch_bounds__(256) void acoef_gemm(
    const float* __restrict__ Qs, const float* __restrict__ Wk,
    float* __restrict__ Acoef) {
  const int wave = threadIdx.x >> 5, lane = threadIdx.x & 31;
  const int t = blockIdx.x * 8 + wave;   // 0..383 (8 heads * 24 * 2)
  const int h = t / 48;
  const int rem = t % 48;
  const int tm = rem >> 1, tn = rem & 1;
  const int r16 = lane & 15;
  const int kh = (lane >> 4) << 1;
  const int arow = tm * 16 + r16;
  const int pcol = tn * 16 + r16;

  v8f acc = {};
  for (int k0 = 0; k0 < DH_; k0 += 4) {
    v2f a = *(const v2f*)(Qs + arow * DIM_ + h * DH_ + k0 + kh);
    // B[k=d][n=p] = Wk[(256+p)*256 + h*32 + d]  (contiguous in d -> v2f)
    v2f b = *(const v2f*)(Wk + (DIM_ + pcol) * DIM_ + h * DH_ + k0 + kh);
    acc = wmma4(a, b, acc);
  }
  const int rbase = (lane >> 4) * 8;
#pragma unroll
  for (int r = 0; r < 8; ++r)
    Acoef[(h * L_ + tm * 16 + rbase + r) * DH_ + pcol] = acc[r];
}

// ---------------------------------------------------------------------------
// Kernel 3: S0[h,i,j] = Qs_h[i,:] . Kb_h[j,:]   (scale already in Qs)
// per head: [384,32] @ [32,384].  grid = 576, block = 256.
// ---------------------------------------------------------------------------
__global__ __launch_bounds__(256) void score_gemm(
    const float* __restrict__ Qs, const float* __restrict__ Kb,
    float* __restrict__ S0) {
  const int wave = threadIdx.x >> 5, lane = threadIdx.x & 31;
  const int t = blockIdx.x * 8 + wave;   // 0..4607 (8 heads * 24 * 24)
  const int h = t / 576;
  const int rem = t % 576;
  const int tm = rem / 24, tn = rem % 24;
  const int r16 = lane & 15;
  const int kh = (lane >> 4) << 1;
  const int irow = tm * 16 + r16;
  const int jcol = tn * 16 + r16;

  v8f acc = {};
  for (int k0 = 0; k0 < DH_; k0 += 4) {
    v2f a = *(const v2f*)(Qs + irow * DIM_ + h * DH_ + k0 + kh);
    // B[k=d][n=j] = Kb[j*256 + h*32 + d]  (contiguous in d -> v2f)
    v2f b = *(const v2f*)(Kb + jcol * DIM_ + h * DH_ + k0 + kh);
    acc = wmma4(a, b, acc);
  }
  const int rbase = (lane >> 4) * 8;
#pragma unroll
  for (int r = 0; r < 8; ++r)
    S0[(h * L_ + tm * 16 + rbase + r) * L_ + tn * 16 + r16] = acc[r];
}

// ---------------------------------------------------------------------------
// Kernel 4: per-query attention core. One block per query i, 256 threads.
// Wave h owns head h for softmax + output reductions.
// ---------------------------------------------------------------------------
__global__ __launch_bounds__(256) void attn_kernel(
    const float* __restrict__ positions, const float* __restrict__ Vb,
    const float* __restrict__ Wv, const float* __restrict__ Acoef,
    const float* __restrict__ S0, float* __restrict__ Ovec) {
  __shared__ float penc[L_ * 32];   // 48 KB: pos_enc(i, j, :)
  __shared__ float sc[H_ * L_];     // 12 KB: scores -> attn weights
  __shared__ float Ai[H_ * 32];     //  1 KB: per-head score coefficients

  const int i = blockIdx.x;
  const int tid = threadIdx.x;

  // Load A[h, i, :] (8 heads * 32 = 256 = blockDim)
  Ai[tid] = Acoef[((tid >> 5) * L_ + i) * DH_ + (tid & 31)];

  const float pxi = positions[i * 2 + 0];
  const float pyi = positions[i * 2 + 1];

  // Positional encoding for all keys j (heavy transcendental part).
  for (int j = tid; j < L_; j += 256) {
    const float dx = pxi - positions[j * 2 + 0];
    const float dy = pyi - positions[j * 2 + 1];
    const float dist = sqrtf(dx * dx + dy * dy + 1e-8f);
    const float dn = dist / 10.0f;
    const float dc = dn / (1.0f + dn);
    const float an = (atan2f(dy, dx) + PI_F) / (2.0f * PI_F);
#pragma unroll
    for (int f = 0; f < 8; ++f) {
      const float freq = 1.0f + (9.0f / 7.0f) * (float)f;
      const float de = dc * freq * PI_F;
      const float ae = an * freq * PI_F;
      penc[j * 32 + f]      = sinf(de);
      penc[j * 32 + 8 + f]  = cosf(de);
      penc[j * 32 + 16 + f] = sinf(ae);
      penc[j * 32 + 24 + f] = cosf(ae);
    }
  }
  __syncthreads();

  // score[h,j] = S0[h,i,j] + penc(i,j,:) . A[h,i,:]
  for (int idx = tid; idx < H_ * L_; idx += 256) {
    const int h = idx / L_;
    const int j = idx - h * L_;
    const float* Ar = &Ai[h * 32];
    const float* pj = &penc[j * 32];
    float dot = S0[(h * L_ + i) * L_ + j];
#pragma unroll
    for (int p = 0; p < 32; ++p) dot += pj[p] * Ar[p];
    sc[h * L_ + j] = dot;
  }
  __syncthreads();

  // Wave h handles head h from here on.
  const int h = tid >> 5;
  const int lane = tid & 31;
  float* srow = &sc[h * L_];

  float m = -1e30f;
  for (int j = lane; j < L_; j += 32) m = fmaxf(m, srow[j]);
#pragma unroll
  for (int o = 16; o > 0; o >>= 1) m = fmaxf(m, __shfl_xor(m, o, 32));

  float sum = 0.0f;
  for (int j = lane; j < L_; j += 32) {
    const float e = expf(srow[j] - m);
    srow[j] = e;
    sum += e;
  }
#pragma unroll
  for (int o = 16; o > 0; o >>= 1) sum += __shfl_xor(sum, o, 32);
  const float inv = 1.0f / sum;
  for (int j = lane; j < L_; j += 32) srow[j] *= inv;

  // out_base[h, lane] = sum_j attn * Vb[j, h*32+lane]   (coalesced 128B/j)
  float ob = 0.0f;
  const float* vcol = Vb + h * DH_ + lane;
  for (int j = 0; j < L_; ++j) ob += srow[j] * vcol[j * DIM_];

  // s[h, lane] = sum_j attn * penc[j, lane]
  float s = 0.0f;
  for (int j = 0; j < L_; ++j) s += srow[j] * penc[j * 32 + lane];

  // out[h, lane] = out_base + sum_p s_p * Wv[256+p, h*32+lane]
  float outv = ob;
#pragma unroll
  for (int p = 0; p < 32; ++p) {
    const float sp = __shfl(s, p, 32);
    outv += sp * Wv[(DIM_ + p) * DIM_ + h * DH_ + lane];
  }
  Ovec[i * DIM_ + h * DH_ + lane] = outv;
}

// ---------------------------------------------------------------------------
// Kernel 5: out = Ovec @ Wo + bo.  grid = 48, block = 256.
// ---------------------------------------------------------------------------
__global__ __launch_bounds__(256) void out_gemm(
    const float* __restrict__ Ovec, const float* __restrict__ Wo,
    const float* __restrict__ bo, float* __restrict__ out) {
  const int wave = threadIdx.x >> 5, lane = threadIdx.x & 31;
  const int t = blockIdx.x * 8 + wave;   // 0..383 tiles
  const int tm = t >> 4, tn = t & 15;
  const int r16 = lane & 15;
  const int kh = (lane >> 4) << 1;
  const int arow = tm * 16 + r16;
  const int bcol = tn * 16 + r16;

  v8f acc = {};
  for (int k0 = 0; k0 < DIM_; k0 += 4) {
    v2f a = *(const v2f*)(Ovec + arow * DIM_ + k0 + kh);
    v2f b;
    b.x = Wo[(k0 + kh) * DIM_ + bcol];
    b.y = Wo[(k0 + kh + 1) * DIM_ + bcol];
    acc = wmma4(a, b, acc);
  }
  const int rbase = (lane >> 4) * 8;
  const float bias = bo[tn * 16 + r16];
#pragma unroll
  for (int r = 0; r < 8; ++r)
    out[(tm * 16 + rbase + r) * DIM_ + tn * 16 + r16] = acc[r] + bias;
}

// ---------------------------------------------------------------------------
extern "C" void kernel_launch(void* const* d_in, const int* in_sizes, int n_in,
                              void* d_out, int out_size, void* d_ws,
                              size_t ws_size, hipStream_t stream) {
  const float* x         = (const float*)d_in[0];
  const float* positions = (const float*)d_in[1];
  const float* Wq        = (const float*)d_in[2];
  const float* Wk        = (const float*)d_in[3];
  const float* Wv        = (const float*)d_in[4];
  const float* Wo        = (const float*)d_in[5];
  const float* bo        = (const float*)d_in[6];
  float* out = (float*)d_out;

  // Workspace layout (floats): total ~6.7 MB
  float* ws    = (float*)d_ws;
  float* Qs    = ws;                 // 384*256
  float* Kb    = Qs + L_ * DIM_;     // 384*256
  float* Vb    = Kb + L_ * DIM_;     // 384*256
  float* Acoef = Vb + L_ * DIM_;     // 8*384*32
  float* Ovec  = Acoef + H_ * L_ * DH_;  // 384*256
  float* S0    = Ovec + L_ * DIM_;   // 8*384*384

  qkv_gemm<<<dim3(48, 3), 256, 0, stream>>>(x, Wq, Wk, Wv, Qs, Kb, Vb);
  acoef_gemm<<<48, 256, 0, stream>>>(Qs, Wk, Acoef);
  score_gemm<<<576, 256, 0, stream>>>(Qs, Kb, S0);
  attn_kernel<<<384, 256, 0, stream>>>(positions, Vb, Wv, Acoef, S0, Ovec);
  out_gemm<<<48, 256, 0, stream>>>(Ovec, Wo, bo, out);
}